// MovementGATModel_83141976916257
// MI455X (gfx1250) — compile-verified
//
#include <hip/hip_runtime.h>

// MovementGATModel for MI455X (gfx1250, wave32).
// Pipeline: CSR build (once, graph shared by both layers) ->
//   [WMMA fp32 GEMM (phi-free depth-2 pipeline) -> per-node attn scores ->
//    segment softmax -> per-dst register-accumulated aggregation ->
//    batchnorm+ReLU] x2 -> fc+mask.
// All buffers L2-resident (~20MB total); ws usage ~17.2MB.

#define NNODES 2560
#define NEDGES 163840
#define TEDGES (NEDGES + NNODES)   // +self loops = 166400
#define DH     512
#define NEG_SLOPE 0.2f

typedef float v2f __attribute__((ext_vector_type(2)));
typedef float v8f __attribute__((ext_vector_type(8)));

__device__ __forceinline__ float wave_sum32(float v) {
  #pragma unroll
  for (int o = 16; o > 0; o >>= 1) v += __shfl_xor(v, o, 32);
  return v;
}
__device__ __forceinline__ float wave_max32(float v) {
  #pragma unroll
  for (int o = 16; o > 0; o >>= 1) v = fmaxf(v, __shfl_xor(v, o, 32));
  return v;
}

// ---------------- CSR build ----------------
__global__ void k_zero_counts(int* __restrict__ counts) {
  int i = blockIdx.x * blockDim.x + threadIdx.x;
  if (i < NNODES) counts[i] = 0;
}

__global__ void k_count(const int* __restrict__ ei, int* __restrict__ counts) {
  int e = blockIdx.x * blockDim.x + threadIdx.x;
  if (e >= TEDGES) return;
  int d = (e < NEDGES) ? ei[NEDGES + e] : (e - NEDGES);
  atomicAdd(&counts[d], 1);
}

// single-wave exclusive scan: 32 lanes x 80 elements each
__global__ void k_scan(const int* __restrict__ counts, int* __restrict__ row_ptr,
                       int* __restrict__ row_cur) {
  const int lane = threadIdx.x;            // 0..31
  const int CH = NNODES / 32;              // 80
  const int base = lane * CH;
  int local = 0;
  for (int i = 0; i < CH; ++i) local += counts[base + i];
  int x = local;
  #pragma unroll
  for (int off = 1; off < 32; off <<= 1) {
    int y = __shfl_up(x, off, 32);
    if (lane >= off) x += y;
  }
  int run = x - local;                     // exclusive prefix of this chunk
  for (int i = 0; i < CH; ++i) {
    row_ptr[base + i] = run;
    row_cur[base + i] = run;
    run += counts[base + i];
  }
  if (lane == 31) row_ptr[NNODES] = run;   // == TEDGES
}

__global__ void k_scatter(const int* __restrict__ ei, int* __restrict__ row_cur,
                          int* __restrict__ csr_src) {
  int e = blockIdx.x * blockDim.x + threadIdx.x;
  if (e >= TEDGES) return;
  int d, s;
  if (e < NEDGES) { s = ei[e]; d = ei[NEDGES + e]; } else { s = e - NEDGES; d = s; }
  int pos = atomicAdd(&row_cur[d], 1);
  csr_src[pos] = s;
}

// per-dst insertion sort by src -> deterministic reduction order every replay
__global__ void k_sort_segments(int* __restrict__ csr_src, const int* __restrict__ row_ptr) {
  int d = blockIdx.x * blockDim.x + threadIdx.x;
  if (d >= NNODES) return;
  int beg = row_ptr[d], end = row_ptr[d + 1];
  for (int i = beg + 1; i < end; ++i) {
    int key = csr_src[i];
    int j = i - 1;
    while (j >= beg && csr_src[j] > key) { csr_src[j + 1] = csr_src[j]; --j; }
    csr_src[j + 1] = key;
  }
}

// ---------------- WMMA GEMM: C[M,Nd] = A[M,K] @ W[Nd,K]^T ----------------
// block = 128 threads (4 waves); wave computes 16(M) x 64(N); grid = (Nd/256, M/16)
// Phi-free depth-2 software pipeline: named registers per parity, unconditional
// prefetch in the steady state, two-batch epilogue. Loads are issued half a loop
// iteration (4 WMMAs) before their consuming s_wait.
#define WMMA4F32(ACC, AV, BV) \
  ACC = __builtin_amdgcn_wmma_f32_16x16x4_f32(false, AV, false, BV, (short)0, ACC, false, false)

template <int K>
__global__ __launch_bounds__(128) void k_gemm_xwT(const float* __restrict__ A,
                                                  const float* __restrict__ W,
                                                  float* __restrict__ C,
                                                  int Nd) {
  const int lane = threadIdx.x & 31;
  const int wave = threadIdx.x >> 5;               // 0..3
  const int l16  = lane & 15;
  const int half = lane >> 4;                      // 0/1
  const int row0 = blockIdx.y * 16;
  const int col0 = blockIdx.x * 256 + wave * 64;

  v8f acc0 = (v8f){0.f,0.f,0.f,0.f,0.f,0.f,0.f,0.f};
  v8f acc1 = acc0, acc2 = acc0, acc3 = acc0;

  // A lane view: A[row0+l16][k + 2*half .. +1]   (16x4 f32 tile layout)
  const float* Ap  = A + (size_t)(row0 + l16) * K + 2 * half;
  // B lane view: B[k][n] = W[n][k] -> W[col+j*16+l16][k + 2*half .. +1]
  const float* Wp0 = W + (size_t)(col0 +  0 + l16) * K + 2 * half;
  const float* Wp1 = W + (size_t)(col0 + 16 + l16) * K + 2 * half;
  const float* Wp2 = W + (size_t)(col0 + 32 + l16) * K + 2 * half;
  const float* Wp3 = W + (size_t)(col0 + 48 + l16) * K + 2 * half;

  constexpr int NB = K / 4;                        // 64 or 128 (even)

  // preload batches 0 (parity 0) and 1 (parity 1)
  v2f a0  = *(const v2f*)(Ap);       v2f a1  = *(const v2f*)(Ap  + 4);
  v2f b00 = *(const v2f*)(Wp0);      v2f b01 = *(const v2f*)(Wp0 + 4);
  v2f b10 = *(const v2f*)(Wp1);      v2f b11 = *(const v2f*)(Wp1 + 4);
  v2f b20 = *(const v2f*)(Wp2);      v2f b21 = *(const v2f*)(Wp2 + 4);
  v2f b30 = *(const v2f*)(Wp3);      v2f b31 = *(const v2f*)(Wp3 + 4);

  #pragma unroll 1
  for (int i = 0; i < NB - 2; i += 2) {
    // consume batch i (parity 0), then refill with batch i+2
    WMMA4F32(acc0, a0, b00);
    WMMA4F32(acc1, a0, b10);
    WMMA4F32(acc2, a0, b20);
    WMMA4F32(acc3, a0, b30);
    {
      const int k = (i + 2) * 4;
      a0  = *(const v2f*)(Ap  + k);
      b00 = *(const v2f*)(Wp0 + k);
      b10 = *(const v2f*)(Wp1 + k);
      b20 = *(const v2f*)(Wp2 + k);
      b30 = *(const v2f*)(Wp3 + k);
    }
    // consume batch i+1 (parity 1), then refill with batch i+3
    WMMA4F32(acc0, a1, b01);
    WMMA4F32(acc1, a1, b11);
    WMMA4F32(acc2, a1, b21);
    WMMA4F32(acc3, a1, b31);
    {
      const int k = (i + 3) * 4;
      a1  = *(const v2f*)(Ap  + k);
      b01 = *(const v2f*)(Wp0 + k);
      b11 = *(const v2f*)(Wp1 + k);
      b21 = *(const v2f*)(Wp2 + k);
      b31 = *(const v2f*)(Wp3 + k);
    }
  }
  // epilogue: batches NB-2 (parity 0) and NB-1 (parity 1), no more loads
  WMMA4F32(acc0, a0, b00);
  WMMA4F32(acc1, a0, b10);
  WMMA4F32(acc2, a0, b20);
  WMMA4F32(acc3, a0, b30);
  WMMA4F32(acc0, a1, b01);
  WMMA4F32(acc1, a1, b11);
  WMMA4F32(acc2, a1, b21);
  WMMA4F32(acc3, a1, b31);

  // D layout: VGPR v -> row = row0 + 8*half + v; col = col0 + 16*j + l16
  float* Cw = C + (size_t)(row0 + half * 8) * Nd + col0 + l16;
  #pragma unroll
  for (int v = 0; v < 8; ++v) {
    Cw[(size_t)v * Nd +  0] = acc0[v];
    Cw[(size_t)v * Nd + 16] = acc1[v];
    Cw[(size_t)v * Nd + 32] = acc2[v];
    Cw[(size_t)v * Nd + 48] = acc3[v];
  }
}

// ---------------- attention scores: s_src[n]=h[n].a_src, s_dst[n]=h[n].a_dst ---
__global__ __launch_bounds__(128) void k_score(const float* __restrict__ h,
                                               const float* __restrict__ a_src,
                                               const float* __restrict__ a_dst,
                                               float* __restrict__ s_src,
                                               float* __restrict__ s_dst) {
  int gid = blockIdx.x * blockDim.x + threadIdx.x;
  int wid = gid >> 5, lane = threadIdx.x & 31;
  if (wid >= NNODES) return;
  const float* hr = h + (size_t)wid * DH;
  float ss = 0.f, sd = 0.f;
  #pragma unroll 4
  for (int t = lane; t < DH; t += 32) {
    float v = hr[t];
    ss = fmaf(v, a_src[t], ss);
    sd = fmaf(v, a_dst[t], sd);
  }
  ss = wave_sum32(ss);
  sd = wave_sum32(sd);
  if (lane == 0) { s_src[wid] = ss; s_dst[wid] = sd; }
}

// ---------------- segment softmax over CSR (one wave per dst) ----------------
__global__ __launch_bounds__(128) void k_softmax(const float* __restrict__ s_src,
                                                 const float* __restrict__ s_dst,
                                                 const int* __restrict__ row_ptr,
                                                 const int* __restrict__ csr_src,
                                                 float* __restrict__ alpha) {
  int gid = blockIdx.x * blockDim.x + threadIdx.x;
  int wid = gid >> 5, lane = threadIdx.x & 31;
  if (wid >= NNODES) return;
  int beg = row_ptr[wid], end = row_ptr[wid + 1];
  float sd = s_dst[wid];
  float m = -3.4e38f;
  for (int i = beg + lane; i < end; i += 32) {
    float l = s_src[csr_src[i]] + sd;
    l = (l > 0.f) ? l : NEG_SLOPE * l;
    m = fmaxf(m, l);
  }
  m = wave_max32(m);
  float sum = 0.f;
  for (int i = beg + lane; i < end; i += 32) {
    float l = s_src[csr_src[i]] + sd;
    l = (l > 0.f) ? l : NEG_SLOPE * l;
    float e = __expf(l - m);
    alpha[i] = e;
    sum += e;
  }
  sum = wave_sum32(sum);
  float rs = 1.f / sum;
  for (int i = beg + lane; i < end; i += 32) alpha[i] *= rs;
}

// ---------------- aggregation: out[d] = sum alpha*h[src] + bias --------------
#define AGG_CHUNK 64
__global__ __launch_bounds__(256) void k_aggregate(const float* __restrict__ h,
                                                   const float* __restrict__ alpha,
                                                   const int* __restrict__ csr_src,
                                                   const int* __restrict__ row_ptr,
                                                   const float* __restrict__ bias,
                                                   float* __restrict__ out) {
  __shared__ int   s_idx[AGG_CHUNK];
  __shared__ float s_al[AGG_CHUNK];
  const int d = blockIdx.x;
  const int t = threadIdx.x;
  const int beg = row_ptr[d], end = row_ptr[d + 1];
  float acc0 = 0.f, acc1 = 0.f;
  for (int base = beg; base < end; base += AGG_CHUNK) {
    int cnt = end - base; if (cnt > AGG_CHUNK) cnt = AGG_CHUNK;
    if (t < cnt) { s_idx[t] = csr_src[base + t]; s_al[t] = alpha[base + t]; }
    __syncthreads();
    for (int j = 0; j < cnt; ++j) {
      const float* hr = h + (size_t)s_idx[j] * DH;
      float a = s_al[j];
      acc0 = fmaf(a, hr[t],       acc0);
      acc1 = fmaf(a, hr[t + 256], acc1);
    }
    __syncthreads();
  }
  out[(size_t)d * DH + t]       = acc0 + bias[t];
  out[(size_t)d * DH + t + 256] = acc1 + bias[t + 256];
}

// ---------------- batchnorm ----------------
__global__ __launch_bounds__(256) void k_bn_stats(const float* __restrict__ x,
                                                  float* __restrict__ mu,
                                                  float* __restrict__ rstd) {
  int c = blockIdx.x * blockDim.x + threadIdx.x;   // < 512
  float s = 0.f, sq = 0.f;
  #pragma unroll 4
  for (int n = 0; n < NNODES; ++n) {
    float v = x[(size_t)n * DH + c];
    s += v;
    sq = fmaf(v, v, sq);
  }
  const float inv = 1.f / (float)NNODES;
  float m = s * inv;
  float var = sq * inv - m * m;
  if (var < 0.f) var = 0.f;
  mu[c] = m;
  rstd[c] = rsqrtf(var + 1e-5f);
}

__global__ __launch_bounds__(256) void k_bn_relu(const float* __restrict__ x,
                                                 const float* __restrict__ gamma,
                                                 const float* __restrict__ beta,
                                                 const float* __restrict__ mu,
                                                 const float* __restrict__ rstd,
                                                 float* __restrict__ y) {
  int i = blockIdx.x * blockDim.x + threadIdx.x;
  if (i >= NNODES * DH) return;
  int c = i & (DH - 1);
  float v = fmaf(gamma[c] * rstd[c], x[i] - mu[c], beta[c]);
  y[i] = (v > 0.f) ? v : 0.f;
}

// ---------------- fc + mask ----------------
__global__ __launch_bounds__(128) void k_fc(const float* __restrict__ h,
                                            const float* __restrict__ fcW,
                                            const float* __restrict__ fcb,
                                            const float* __restrict__ mask,
                                            float* __restrict__ out) {
  int gid = blockIdx.x * blockDim.x + threadIdx.x;
  int wid = gid >> 5, lane = threadIdx.x & 31;
  if (wid >= NNODES) return;
  const float* hr = h + (size_t)wid * DH;
  float d0 = 0.f, d1 = 0.f;
  #pragma unroll 4
  for (int t = lane; t < DH; t += 32) {
    float v = hr[t];
    d0 = fmaf(v, fcW[t],      d0);
    d1 = fmaf(v, fcW[DH + t], d1);
  }
  d0 = wave_sum32(d0);
  d1 = wave_sum32(d1);
  if (lane == 0) {
    float mk = mask[wid];
    out[(size_t)wid * 2 + 0] = (d0 + fcb[0]) * mk;
    out[(size_t)wid * 2 + 1] = (d1 + fcb[1]) * mk;
  }
}

extern "C" void kernel_launch(void* const* d_in, const int* in_sizes, int n_in,
                              void* d_out, int out_size, void* d_ws, size_t ws_size,
                              hipStream_t stream) {
  (void)in_sizes; (void)n_in; (void)out_size; (void)ws_size;
  const float* x      = (const float*)d_in[0];
  const int*   ei     = (const int*)  d_in[1];
  const float* mask   = (const float*)d_in[2];
  const float* W1     = (const float*)d_in[3];
  const float* asrc1  = (const float*)d_in[4];
  const float* adst1  = (const float*)d_in[5];
  const float* b1     = (const float*)d_in[6];
  const float* g1     = (const float*)d_in[7];
  const float* be1    = (const float*)d_in[8];
  const float* W2     = (const float*)d_in[9];
  const float* asrc2  = (const float*)d_in[10];
  const float* adst2  = (const float*)d_in[11];
  const float* b2     = (const float*)d_in[12];
  const float* g2     = (const float*)d_in[13];
  const float* be2    = (const float*)d_in[14];
  const float* fcW    = (const float*)d_in[15];
  const float* fcb    = (const float*)d_in[16];
  float* out = (float*)d_out;

  // workspace carve-up (~17.2 MB)
  float* h     = (float*)d_ws;                    // 2560*512
  float* aggo  = h    + (size_t)NNODES * DH;      // 2560*512
  float* act   = aggo + (size_t)NNODES * DH;      // 2560*512
  float* ssrc  = act  + (size_t)NNODES * DH;      // 2560
  float* sdst  = ssrc + NNODES;                   // 2560
  float* mu    = sdst + NNODES;                   // 512
  float* rstd  = mu   + DH;                       // 512
  float* alpha = rstd + DH;                       // TEDGES
  int* row_ptr = (int*)(alpha + TEDGES);          // NNODES+1
  int* row_cur = row_ptr + (NNODES + 1);          // NNODES
  int* counts  = row_cur + NNODES;                // NNODES
  int* csr_src = counts  + NNODES;                // TEDGES

  const int eg = (TEDGES + 255) / 256;
  const int ng = (NNODES + 255) / 256;

  // CSR build (graph shared by both GAT layers)
  k_zero_counts<<<ng, 256, 0, stream>>>(counts);
  k_count<<<eg, 256, 0, stream>>>(ei, counts);
  k_scan<<<1, 32, 0, stream>>>(counts, row_ptr, row_cur);
  k_scatter<<<eg, 256, 0, stream>>>(ei, row_cur, csr_src);
  k_sort_segments<<<ng, 256, 0, stream>>>(csr_src, row_ptr);

  dim3 ggrid(DH / 256, NNODES / 16);              // (2, 160)
  const int wgrid = NNODES / 4;                   // 4 waves/block of 128

  // ---- layer 1 ----
  k_gemm_xwT<256><<<ggrid, 128, 0, stream>>>(x, W1, h, DH);
  k_score<<<wgrid, 128, 0, stream>>>(h, asrc1, adst1, ssrc, sdst);
  k_softmax<<<wgrid, 128, 0, stream>>>(ssrc, sdst, row_ptr, csr_src, alpha);
  k_aggregate<<<NNODES, 256, 0, stream>>>(h, alpha, csr_src, row_ptr, b1, aggo);
  k_bn_stats<<<2, 256, 0, stream>>>(aggo, mu, rstd);
  k_bn_relu<<<(NNODES * DH) / 256, 256, 0, stream>>>(aggo, g1, be1, mu, rstd, act);

  // ---- layer 2 ----
  k_gemm_xwT<512><<<ggrid, 128, 0, stream>>>(act, W2, h, DH);
  k_score<<<wgrid, 128, 0, stream>>>(h, asrc2, adst2, ssrc, sdst);
  k_softmax<<<wgrid, 128, 0, stream>>>(ssrc, sdst, row_ptr, csr_src, alpha);
  k_aggregate<<<NNODES, 256, 0, stream>>>(h, alpha, csr_src, row_ptr, b2, aggo);
  k_bn_stats<<<2, 256, 0, stream>>>(aggo, mu, rstd);
  k_bn_relu<<<(NNODES * DH) / 256, 256, 0, stream>>>(aggo, g2, be2, mu, rstd, act);

  // ---- head ----
  k_fc<<<wgrid, 128, 0, stream>>>(act, fcW, fcb, mask, out);
}